// LearnedSortOrder_v2_34376918237594
// MI455X (gfx1250) — compile-verified
//
#include <hip/hip_runtime.h>
#include <stdint.h>

// ---------------------------------------------------------------------------
// LearnedSortOrder on MI455X (gfx1250, wave32).
//
// Cost model: output = 2 * N * 64 f32 = 512 MB of streaming stores -> ~22us
// floor at 23.3 TB/s. MLP math (~128 MFLOP) is free. So: tiny WMMA-based
// score/minmax pass, tiny bin-index pass (1 B/elem), then one flat float4
// non-temporal store kernel for the one-hot.
// ---------------------------------------------------------------------------

typedef __attribute__((ext_vector_type(2))) float v2f;
typedef __attribute__((ext_vector_type(4))) float v4f;
typedef __attribute__((ext_vector_type(8))) float v8f;

#define BLK 256          // 8 waves per workgroup (wave32)
#define RED_BLOCKS 256   // grid for the reduction/index passes

// ---------------------------------------------------------------------------
// Per-lane preloaded MLP weights for the WMMA K-loop.
// K index for this lane at WMMA step s:  k0(s) = 4*s + (lane/16)*2, k1 = k0+1
// (matches the f32 16x4 A layout: lanes 0-15 hold K={0,1}, lanes 16-31 K={2,3})
// ---------------------------------------------------------------------------
struct LaneW {
  float w1a[8], w1b[8];
  float b1a[8], b1b[8];
  float w2a[8], w2b[8];
  float bias2;
};

__device__ __forceinline__ void preload_lanew(LaneW& lw, const float* W1,
                                              const float* b1, const float* W2,
                                              const float* b2, int lane) {
  const int koff = (lane >> 4) << 1;  // 0 or 2
#pragma unroll
  for (int s = 0; s < 8; ++s) {
    const int k0 = 4 * s + koff;
    lw.w1a[s] = W1[k0];     lw.w1b[s] = W1[k0 + 1];
    lw.b1a[s] = b1[k0];     lw.b1b[s] = b1[k0 + 1];
    lw.w2a[s] = W2[k0];     lw.w2b[s] = W2[k0 + 1];
  }
  lw.bias2 = b2[0];
}

// ---------------------------------------------------------------------------
// Compute scores for one 16-element tile with v_wmma_f32_16x16x4_f32.
//   A[16x4] step s: A[m][k] = relu(x[tile*16+m] * W1[k] + b1[k]), k = 4s..4s+3
//   B[4x16]       : every column = W2[4s..4s+3]  (so every D column = scores)
//   D[16x16] accumulates over 8 steps -> D[m][*] = (h @ W2)[m]
// D layout: lane L holds rows M = 8*(L/16) + v (v = element 0..7), any column.
// Return: score for element  melem(L) = 8*(L/16) + (L&7); each score appears
// in two lanes (L and L^8) -> full-wave min/max reduction needs no masking.
// ---------------------------------------------------------------------------
__device__ __forceinline__ float tile_score(const LaneW& lw, const float* x,
                                            int N, int tile, int lane) {
  const int m = lane & 15;                 // A-matrix row handled by this lane
  int e = tile * 16 + m;
  if (e >= N) e = N - 1;                   // clamp (N % 16 == 0 for N=1e6)
  const float xm = x[e];

  v8f c = {};
#pragma unroll
  for (int s = 0; s < 8; ++s) {
    v2f a, b;
    a.x = fmaxf(fmaf(xm, lw.w1a[s], lw.b1a[s]), 0.0f);
    a.y = fmaxf(fmaf(xm, lw.w1b[s], lw.b1b[s]), 0.0f);
    b.x = lw.w2a[s];
    b.y = lw.w2b[s];
    // 8 args: (neg_a, A, neg_b, B, c_mod, C, reuse_a, reuse_b)
    c = __builtin_amdgcn_wmma_f32_16x16x4_f32(false, a, false, b, (short)0, c,
                                              false, false);
  }

  // Extract D element (lane&7) -> branchless select chain.
  const int v = lane & 7;
  float acc = c[0];
  acc = (v == 1) ? c[1] : acc;
  acc = (v == 2) ? c[2] : acc;
  acc = (v == 3) ? c[3] : acc;
  acc = (v == 4) ? c[4] : acc;
  acc = (v == 5) ? c[5] : acc;
  acc = (v == 6) ? c[6] : acc;
  acc = (v == 7) ? c[7] : acc;

  return fmaxf(acc + lw.bias2, 0.0f);      // second ReLU
}

// ---------------------------------------------------------------------------
// Kernel 0: reset min/max control cells (run every launch; ws is not
// re-poisoned between graph replays, but we re-init deterministically).
// ---------------------------------------------------------------------------
__global__ void init_ctrl_kernel(unsigned* ctrl) {
  ctrl[0] = 0x7F800000u;  // +inf bits  (min identity; scores >= 0)
  ctrl[1] = 0x00000000u;  // 0.0f bits  (max identity; scores >= 0)
}

// ---------------------------------------------------------------------------
// Kernel 1: scores + global min/max (uint-ordered atomics, 2 per block).
// ---------------------------------------------------------------------------
__global__ void score_minmax_kernel(const float* __restrict__ x,
                                    const float* __restrict__ W1,
                                    const float* __restrict__ b1,
                                    const float* __restrict__ W2,
                                    const float* __restrict__ b2,
                                    unsigned* __restrict__ ctrl, int N) {
  const int lane = threadIdx.x & 31;
  LaneW lw;
  preload_lanew(lw, W1, b1, W2, b2, lane);

  const int wid = (blockIdx.x * blockDim.x + threadIdx.x) >> 5;  // wave id
  const int nwaves = (gridDim.x * blockDim.x) >> 5;
  const int numTiles = (N + 15) >> 4;

  float tmin = __builtin_huge_valf();
  float tmax = 0.0f;
  for (int t = wid; t < numTiles; t += nwaves) {   // uniform per wave: EXEC=~0
    const float sc = tile_score(lw, x, N, t, lane);
    tmin = fminf(tmin, sc);
    tmax = fmaxf(tmax, sc);
  }

  // wave32 butterfly reduction
#pragma unroll
  for (int off = 16; off > 0; off >>= 1) {
    tmin = fminf(tmin, __shfl_xor(tmin, off, 32));
    tmax = fmaxf(tmax, __shfl_xor(tmax, off, 32));
  }

  __shared__ float smin[BLK / 32], smax[BLK / 32];
  const int w = threadIdx.x >> 5;
  if (lane == 0) { smin[w] = tmin; smax[w] = tmax; }
  __syncthreads();
  if (threadIdx.x == 0) {
    float bm = smin[0], bM = smax[0];
#pragma unroll
    for (int i = 1; i < BLK / 32; ++i) {
      bm = fminf(bm, smin[i]);
      bM = fmaxf(bM, smax[i]);
    }
    // non-negative floats: bit pattern order == numeric order
    atomicMin(&ctrl[0], __float_as_uint(bm));
    atomicMax(&ctrl[1], __float_as_uint(bM));
  }
}

// ---------------------------------------------------------------------------
// Kernel 2: recompute scores, searchsorted(left) over linspace boundaries,
// store 1-byte bin index. boundaries[k] = s_min + k*step, step=(s_max+1-s_min)/63.
// ---------------------------------------------------------------------------
__global__ void bin_index_kernel(const float* __restrict__ x,
                                 const float* __restrict__ W1,
                                 const float* __restrict__ b1,
                                 const float* __restrict__ W2,
                                 const float* __restrict__ b2,
                                 const unsigned* __restrict__ ctrl,
                                 uint8_t* __restrict__ binidx, int N) {
  const int lane = threadIdx.x & 31;
  LaneW lw;
  preload_lanew(lw, W1, b1, W2, b2, lane);

  const float s_min = floorf(__uint_as_float(ctrl[0]));
  const float s_max = floorf(__uint_as_float(ctrl[1]));
  const float step = (s_max + 1.0f - s_min) * (1.0f / 63.0f);

  const int wid = (blockIdx.x * blockDim.x + threadIdx.x) >> 5;
  const int nwaves = (gridDim.x * blockDim.x) >> 5;
  const int numTiles = (N + 15) >> 4;

  const bool writer = (lane & 15) < 8;                 // 16 unique scores/tile
  const int melem = ((lane >> 4) << 3) | (lane & 7);   // element within tile

  for (int t = wid; t < numTiles; t += nwaves) {
    const float sc = tile_score(lw, x, N, t, lane);
    // first k in [0,64) with boundaries[k] >= sc  (side='left'); may be 64
    int lo = 0, hi = 64;
#pragma unroll
    for (int it = 0; it < 6; ++it) {
      const int mid = (lo + hi) >> 1;
      const float bv = fmaf((float)mid, step, s_min);
      if (bv < sc) lo = mid + 1; else hi = mid;
    }
    const int e = t * 16 + melem;
    if (writer && e < N) binidx[e] = (uint8_t)lo;      // 64 -> all-zero row
  }
}

// ---------------------------------------------------------------------------
// Kernel 3: the bandwidth kernel. One float4 NT store per thread over all
// 2*N*64 output floats (both tuple halves). Perfectly coalesced streaming.
// ---------------------------------------------------------------------------
__global__ void onehot_write_kernel(const uint8_t* __restrict__ binidx,
                                    float* __restrict__ out, int N,
                                    long long total4) {
  const long long g = (long long)blockIdx.x * blockDim.x + threadIdx.x;
  if (g >= total4) return;
  const int row = (int)(g >> 4);              // one-hot row (0 .. 2N-1)
  const int e = (row >= N) ? row - N : row;   // second half duplicates first
  const int c0 = ((int)g & 15) << 2;          // first column of this float4
  const int k = (int)binidx[e];

  v4f val;
  val.x = (k == c0)     ? 1.0f : 0.0f;
  val.y = (k == c0 + 1) ? 1.0f : 0.0f;
  val.z = (k == c0 + 2) ? 1.0f : 0.0f;
  val.w = (k == c0 + 3) ? 1.0f : 0.0f;

  __builtin_nontemporal_store(val, (v4f*)out + g);
}

// ---------------------------------------------------------------------------
extern "C" void kernel_launch(void* const* d_in, const int* in_sizes, int n_in,
                              void* d_out, int out_size, void* d_ws,
                              size_t ws_size, hipStream_t stream) {
  const float* x  = (const float*)d_in[0];
  const float* W1 = (const float*)d_in[1];
  const float* b1 = (const float*)d_in[2];
  const float* W2 = (const float*)d_in[3];
  const float* b2 = (const float*)d_in[4];
  const int N = in_sizes[0];

  // workspace layout: [0,8) = {minbits, maxbits}; [256, 256+N) = uint8 bins
  unsigned* ctrl = (unsigned*)d_ws;
  uint8_t* binidx = (uint8_t*)d_ws + 256;
  float* out = (float*)d_out;

  init_ctrl_kernel<<<1, 1, 0, stream>>>(ctrl);
  score_minmax_kernel<<<RED_BLOCKS, BLK, 0, stream>>>(x, W1, b1, W2, b2, ctrl, N);
  bin_index_kernel<<<RED_BLOCKS, BLK, 0, stream>>>(x, W1, b1, W2, b2, ctrl,
                                                   binidx, N);

  const long long total4 = (long long)out_size >> 2;  // float4 count
  const int blocks3 = (int)((total4 + BLK - 1) / BLK);
  onehot_write_kernel<<<blocks3, BLK, 0, stream>>>(binidx, out, N, total4);
}